// Pointnet2_30657476559126
// MI455X (gfx1250) — compile-verified
//
#include <hip/hip_runtime.h>
#include <hip/hip_bf16.h>

// ---------------------------------------------------------------------------
// PointNet++ (SSG) forward for MI455X / gfx1250, wave32.
//  - fp32 WMMA (v_wmma_f32_16x16x4_f32) for every GEMM (MLP 1x1 convs + FC)
//  - TDM (tensor_load_to_lds) double-buffers K x 16 weight panels in LDS,
//    synced with s_wait_tensorcnt 1/0 (DMA overlapped with WMMA)
//  - All GEMM inputs K-padded to a multiple of 4 -> branch-free A loads
//  - LDS-resident FPS distance array, wave32 ballot ball-query
// ---------------------------------------------------------------------------

typedef float v2f __attribute__((ext_vector_type(2)));
typedef float v8f __attribute__((ext_vector_type(8)));
typedef unsigned int u32;
typedef u32 v4u __attribute__((ext_vector_type(4)));
typedef int  v4i __attribute__((ext_vector_type(4)));
typedef int  v8i __attribute__((ext_vector_type(8)));

#define GEMM_KSTAGE 256   // K rows staged per TDM transfer (16KB per panel)
#define BN_EPS 1e-5f

// ---------------------------------------------------------------------------
// TDM: DMA a tile_w x tile_h fp32 tile (row stride = row_stride elements) from
// global into LDS at lds_byte_addr. Descriptor bitfields per CDNA5 ISA §8.
// tensor_w/tensor_h = remaining in-bounds extents (OOB reads zero-fill).
// ---------------------------------------------------------------------------
__device__ __forceinline__ void tdm_load_2d_f32(u32 lds_byte_addr, const void* gptr,
                                                u32 tile_w, u32 tile_h,
                                                u32 tensor_w, u32 tensor_h,
                                                u32 row_stride)
{
    unsigned long long ga = (unsigned long long)(size_t)gptr;
    v4u g0;
    g0.x = 1u;                                             // count=1, user D#
    g0.y = lds_byte_addr;                                  // lds_addr [63:32]
    g0.z = (u32)ga;                                        // global_addr lo
    g0.w = (u32)((ga >> 32) & 0x1ffffffull) | (2u << 30);  // global_addr hi | type=2
    u32 d0 = (2u << 16);                                   // data_size=2 (4B)
    u32 d1 = (tensor_w & 0xffffu) << 16;                   // tensor_dim0 lo
    u32 d2 = (tensor_w >> 16) | ((tensor_h & 0xffffu) << 16); // dim0 hi | dim1 lo
    u32 d3 = (tensor_h >> 16) | ((tile_w & 0xffffu) << 16);   // dim1 hi | tile_dim0
    u32 d4 = (tile_h & 0xffffu);                           // tile_dim1
    u32 d5 = row_stride;                                   // tensor_dim0_stride lo32
    v8i g1 = { (int)d0, (int)d1, (int)d2, (int)d3, (int)d4, (int)d5, 0, 0 };
    v4i z4 = { 0, 0, 0, 0 };
    v8i z8 = { 0, 0, 0, 0, 0, 0, 0, 0 };
    __builtin_amdgcn_tensor_load_to_lds(g0, g1, z4, z4, z8, 0);
}

// ---------------------------------------------------------------------------
// GEMM: Z(MxN) = X(M x Kpad) @ W(Kw x N) [+ bias], optional per-column
// sum / sum-of-squares accumulation (batch-norm statistics) via atomics.
// One wave per 16x16 output tile; K consumed 4/step with wmma_f32_16x16x4.
// X rows are Kpad-pitched with zero pad columns (Kpad = roundup4(Kw)); the
// W K-tail rows are zero-filled by TDM OOB handling (tensor_h = Kw - k0).
// A-row index is clamped, not masked: OOB A rows only feed OOB D rows.
// ---------------------------------------------------------------------------
__global__ void pn2_gemm_kernel(const float* __restrict__ X,
                                const float* __restrict__ W,
                                const float* __restrict__ bias,
                                float* __restrict__ Z,
                                float* __restrict__ colsum,
                                float* __restrict__ colsumsq,
                                int M, int Kpad, int Kw, int N, int doStats)
{
    extern __shared__ float lb[];           // 2 x (GEMM_KSTAGE x 16) panels
    const int lane = threadIdx.x & 31;
    const int half = lane >> 4;             // 0: lanes 0-15, 1: lanes 16-31
    const int l15  = lane & 15;
    const int m0 = blockIdx.x * 16;
    const int n0 = blockIdx.y * 16;
    int rowc = m0 + l15; if (rowc > M - 1) rowc = M - 1;   // clamp, no mask
    const int col = n0 + l15;
    // LDS generic address low 32 bits == LDS byte offset (ISA §10.2)
    const u32 lds_base = (u32)(size_t)&lb[0];
    const int nstage = (Kpad + GEMM_KSTAGE - 1) / GEMM_KSTAGE;

    // prologue: stage 0 into panel 0
    {
        int kc0 = Kpad < GEMM_KSTAGE ? Kpad : GEMM_KSTAGE;
        tdm_load_2d_f32(lds_base, W + n0, 16u, (u32)kc0,
                        (u32)(N - n0), (u32)Kw, (u32)N);
    }

    v8f c = {};
    const float* Xr = X + (size_t)rowc * Kpad + 2 * half;  // A fragment base
    for (int s = 0; s < nstage; ++s) {
        int k0 = s * GEMM_KSTAGE;
        int kc = Kpad - k0; if (kc > GEMM_KSTAGE) kc = GEMM_KSTAGE;
        if (s + 1 < nstage) {                // issue next stage, wait for older
            int k1 = k0 + GEMM_KSTAGE;
            int kc1 = Kpad - k1; if (kc1 > GEMM_KSTAGE) kc1 = GEMM_KSTAGE;
            tdm_load_2d_f32(lds_base + (u32)((s + 1) & 1) * (GEMM_KSTAGE * 16 * 4),
                            W + (size_t)k1 * N + n0, 16u, (u32)kc1,
                            (u32)(N - n0), (u32)(Kw - k1), (u32)N);
            __builtin_amdgcn_s_wait_tensorcnt((short)1);
        } else {
            __builtin_amdgcn_s_wait_tensorcnt((short)0);
        }
        const float* lp = lb + (size_t)(s & 1) * (GEMM_KSTAGE * 16);
        for (int kk = 0; kk < kc; kk += 4) {
            // A 16x4 fragment: one unconditional 8B load (K is 4-padded)
            v2f a = *(const v2f*)(Xr + k0 + kk);
            // B 4x16 fragment from LDS panel
            v2f b;
            b.x = lp[(kk + half) * 16 + l15];
            b.y = lp[(kk + 2 + half) * 16 + l15];
            c = __builtin_amdgcn_wmma_f32_16x16x4_f32(false, a, false, b,
                                                      (short)0, c, false, false);
        }
    }

    float bv = 0.f;
    if (bias && col < N) bv = bias[col];
    float s1 = 0.f, s2 = 0.f;
#pragma unroll
    for (int i = 0; i < 8; ++i) {           // C/D: VGPR i -> row m0+i+8*half
        int r = m0 + i + half * 8;
        if (r < M && col < N) {
            float v = c[i] + bv;
            Z[(size_t)r * N + col] = v;
            s1 += v; s2 += v * v;
        }
    }
    if (doStats && col < N) {
        atomicAdd(&colsum[col], s1);
        atomicAdd(&colsumsq[col], s2);
    }
}

// ---------------------------------------------------------------------------
// BatchNorm (batch statistics over M rows) + optional ReLU, in place.
// ---------------------------------------------------------------------------
__global__ void pn2_bn_relu_kernel(float* __restrict__ Z,
                                   const float* __restrict__ colsum,
                                   const float* __restrict__ colsumsq,
                                   const float* __restrict__ gamma,
                                   const float* __restrict__ beta,
                                   int M, int N, int relu)
{
    long e = (long)blockIdx.x * blockDim.x + threadIdx.x;
    long total = (long)M * N;
    if (e >= total) return;
    int c = (int)(e % N);
    float inv_m = 1.0f / (float)M;
    float mean = colsum[c] * inv_m;
    float var  = colsumsq[c] * inv_m - mean * mean;
    float v = (Z[e] - mean) * rsqrtf(var + BN_EPS) * gamma[c] + beta[c];
    if (relu) v = fmaxf(v, 0.f);
    Z[e] = v;
}

// ---------------------------------------------------------------------------
// Farthest point sampling. One block per batch; dist[N] in LDS (64KB for
// N=16384, within the 320KB/WGP budget). jnp.argmax first-max tie-breaking.
// ---------------------------------------------------------------------------
__global__ void pn2_fps_kernel(const float* __restrict__ xyz, int N, int S,
                               int* __restrict__ out)
{
    extern __shared__ float fshm[];
    float* dist = fshm;                         // N
    float* rval = fshm + N;                     // blockDim
    int*   ridx = (int*)(rval + blockDim.x);    // blockDim
    __shared__ int s_last;

    int b = blockIdx.x, tid = threadIdx.x;
    const float* P = xyz + (size_t)b * N * 3;
    for (int i = tid; i < N; i += blockDim.x) dist[i] = 1e10f;
    if (tid == 0) { s_last = 0; out[(size_t)b * S] = 0; }
    __syncthreads();

    for (int it = 1; it < S; ++it) {
        int last = s_last;
        float px = P[last * 3], py = P[last * 3 + 1], pz = P[last * 3 + 2];
        float bestv = -1.f; int besti = 0;
        for (int i = tid; i < N; i += blockDim.x) {
            float dx = P[i * 3] - px, dy = P[i * 3 + 1] - py, dz = P[i * 3 + 2] - pz;
            float d = dx * dx + dy * dy + dz * dz;
            float dm = fminf(dist[i], d);
            dist[i] = dm;
            if (dm > bestv) { bestv = dm; besti = i; }   // strict > keeps first max
        }
        rval[tid] = bestv; ridx[tid] = besti;
        __syncthreads();
        for (int s2 = blockDim.x >> 1; s2 > 0; s2 >>= 1) {
            if (tid < s2) {
                float v2 = rval[tid + s2]; int i2 = ridx[tid + s2];
                if (v2 > rval[tid] || (v2 == rval[tid] && i2 < ridx[tid])) {
                    rval[tid] = v2; ridx[tid] = i2;
                }
            }
            __syncthreads();
        }
        if (tid == 0) { s_last = ridx[0]; out[(size_t)b * S + it] = ridx[0]; }
        __syncthreads();
    }
}

// ---------------------------------------------------------------------------
// Ball query: one wave32 per query center; ordered compaction via ballot.
// ---------------------------------------------------------------------------
__global__ void pn2_ballquery_kernel(const float* __restrict__ nxyz,
                                     const float* __restrict__ xyz,
                                     int* __restrict__ nidx,
                                     int B, int S, int N, int ns, float r2)
{
    int wave = blockIdx.x;                    // one 32-thread block per center
    int lane = threadIdx.x & 31;
    if (wave >= B * S) return;
    int b = wave / S;
    const float* ctr = nxyz + (size_t)wave * 3;
    float cx = ctr[0], cy = ctr[1], cz = ctr[2];
    const float* P = xyz + (size_t)b * N * 3;
    int* out = nidx + (size_t)wave * ns;

    int cnt = 0, first = -1;
    for (int base = 0; base < N && cnt < ns; base += 32) {
        int i = base + lane;
        bool ok = false;
        if (i < N) {
            float dx = P[i * 3] - cx, dy = P[i * 3 + 1] - cy, dz = P[i * 3 + 2] - cz;
            ok = (dx * dx + dy * dy + dz * dz) <= r2;
        }
        unsigned mask = (unsigned)__ballot(ok);
        if (first < 0 && mask) first = base + __builtin_ctz(mask);
        int rank = __popc(mask & ((1u << lane) - 1u));
        if (ok && (cnt + rank) < ns) out[cnt + rank] = i;
        cnt += __popc(mask);
    }
    if (cnt > ns) cnt = ns;
    if (first < 0) first = 0;
    for (int j = cnt + lane; j < ns; j += 32) out[j] = first;
}

// ---------------------------------------------------------------------------
// Grouped matrix G (B*S*ns, roundup4(3+C)): cols 0..2 = gathered xyz - center,
// cols 3..3+C-1 = gathered features, pad cols = 0.
// ---------------------------------------------------------------------------
__global__ void pn2_group_kernel(const float* __restrict__ xyz,
                                 const float* __restrict__ feats,
                                 const float* __restrict__ nxyz,
                                 const int* __restrict__ nidx,
                                 float* __restrict__ G,
                                 int B, int S, int N, int ns, int C)
{
    int Cin  = 3 + C;
    int Cin4 = (Cin + 3) & ~3;
    long e = (long)blockIdx.x * blockDim.x + threadIdx.x;
    long total = (long)B * S * ns * Cin4;
    if (e >= total) return;
    int c = (int)(e % Cin4);
    if (c >= Cin) { G[e] = 0.f; return; }    // zero pad column
    long row = e / Cin4;
    int k = (int)(row % ns);
    long bs = row / ns;
    int s = (int)(bs % S);
    int b = (int)(bs / S);
    int id = nidx[((long)b * S + s) * ns + k];
    if (c < 3)
        G[e] = xyz[((long)b * N + id) * 3 + c] - nxyz[((long)b * S + s) * 3 + c];
    else
        G[e] = feats[((long)b * N + id) * C + (c - 3)];
}

__global__ void pn2_gather_points_kernel(float* __restrict__ dst,
                                         const float* __restrict__ src,
                                         const int* __restrict__ idx,
                                         int B, int S, int N, int C)
{
    long e = (long)blockIdx.x * blockDim.x + threadIdx.x;
    long total = (long)B * S * C;
    if (e >= total) return;
    int c = (int)(e % C);
    long bs = e / C;
    int s = (int)(bs % S);
    int b = (int)(bs / S);
    dst[e] = src[((long)b * N + idx[(long)b * S + s]) * C + c];
}

// out[r,c] = max_k X[r,k,c]
__global__ void pn2_maxpool_kernel(const float* __restrict__ X,
                                   float* __restrict__ out,
                                   long R, int Ks, int C)
{
    long e = (long)blockIdx.x * blockDim.x + threadIdx.x;
    if (e >= R * (long)C) return;
    long r = e / C; int c = (int)(e % C);
    const float* p = X + (r * Ks) * (long)C + c;
    float m = p[0];
    for (int k = 1; k < Ks; ++k) m = fmaxf(m, p[(long)k * C]);
    out[r * (long)C + c] = m;
}

__global__ void pn2_copy2d_kernel(float* __restrict__ dst, long dpitch,
                                  const float* __restrict__ src, long spitch,
                                  long rows, long cols)
{
    long e = (long)blockIdx.x * blockDim.x + threadIdx.x;
    if (e >= rows * cols) return;
    long r = e / cols, c = e % cols;
    dst[r * dpitch + c] = src[r * spitch + c];
}

__global__ void pn2_zero_kernel(float* __restrict__ p, long n)
{
    long e = (long)blockIdx.x * blockDim.x + threadIdx.x;
    if (e < n) p[e] = 0.f;
}

// ---------------------------------------------------------------------------
// Host orchestration
// ---------------------------------------------------------------------------
extern "C" void kernel_launch(void* const* d_in, const int* in_sizes, int n_in,
                              void* d_out, int out_size, void* d_ws, size_t ws_size,
                              hipStream_t stream)
{
    const int B = 8, N0 = 16384;

    // ---- input walk: setup_inputs() insertion order ----
    int q = 0;
    const float* pc = (const float*)d_in[q++];   // (8,16384,3)
    struct Lyr { const float *W, *gamma, *beta; };
    Lyr sa[5][3];
    for (int s = 0; s < 5; ++s)
        for (int l = 0; l < 3; ++l) {
            sa[s][l].W     = (const float*)d_in[q++];
            sa[s][l].gamma = (const float*)d_in[q++];
            sa[s][l].beta  = (const float*)d_in[q++];
        }
    const float* w1  = (const float*)d_in[q++];
    const float* b1  = (const float*)d_in[q++];
    const float* g1  = (const float*)d_in[q++];
    const float* be1 = (const float*)d_in[q++];
    const float* w2  = (const float*)d_in[q++];
    const float* b2  = (const float*)d_in[q++];
    const float* g2  = (const float*)d_in[q++];
    const float* be2 = (const float*)d_in[q++];
    const float* w3  = (const float*)d_in[q++];
    const float* b3  = (const float*)d_in[q++];

    // ---- workspace bump allocator (~72 MB) ----
    char* wsp = (char*)d_ws;
    auto alloc = [&](size_t bytes) -> void* {
        void* r = (void*)wsp;
        wsp += (bytes + 255) & ~(size_t)255;
        return r;
    };
    int*   idx1   = (int*)  alloc((size_t)B * 512 * sizeof(int));
    float* xyz1   = (float*)alloc((size_t)B * 512 * 3 * sizeof(float));
    int*   nidx1  = (int*)  alloc((size_t)B * 512 * 32 * sizeof(int));
    int*   idx2   = (int*)  alloc((size_t)B * 256 * sizeof(int));
    float* xyz2   = (float*)alloc((size_t)B * 256 * 3 * sizeof(float));
    int*   nidx2  = (int*)  alloc((size_t)B * 256 * 16 * sizeof(int));
    int*   idx3   = (int*)  alloc((size_t)B * 128 * sizeof(int));
    float* xyz3   = (float*)alloc((size_t)B * 128 * 3 * sizeof(float));
    int*   nidx3  = (int*)  alloc((size_t)B * 128 * 16 * sizeof(int));
    int*   idxs   = (int*)  alloc((size_t)B * 4 * sizeof(int));
    float* sxyz   = (float*)alloc((size_t)B * 4 * 3 * sizeof(float));
    float* Gs     = (float*)alloc((size_t)B * 4 * 4 * sizeof(float));   // padded sep G
    float* feats1 = (float*)alloc((size_t)B * 512 * 64 * sizeof(float));
    float* feats2 = (float*)alloc((size_t)B * 256 * 256 * sizeof(float));
    float* feats3 = (float*)alloc((size_t)B * 128 * 512 * sizeof(float));
    float* feat4  = (float*)alloc((size_t)B * 1024 * sizeof(float));
    float* sep    = (float*)alloc((size_t)B * 1024 * sizeof(float));
    float* stats  = (float*)alloc((size_t)4096 * sizeof(float));  // sum | sumsq
    float* bufA   = (float*)alloc((size_t)8 * 1024 * 1024 * sizeof(float));
    float* bufB   = (float*)alloc((size_t)8 * 1024 * 1024 * sizeof(float));
    float* x1     = (float*)alloc((size_t)B * 2048 * sizeof(float));
    float* h1     = (float*)alloc((size_t)B * 1024 * sizeof(float));
    float* x2     = (float*)alloc((size_t)B * 2048 * sizeof(float));
    float* h2     = (float*)alloc((size_t)B * 2048 * sizeof(float));

    auto nblk = [](long n) { return (unsigned)((n + 255) / 256); };
    auto zero = [&](float* p, long n) {
        pn2_zero_kernel<<<nblk(n), 256, 0, stream>>>(p, n);
    };
    // X must be pitched to roundup4(Kw) with zero pad columns.
    auto gemm = [&](const float* X, const float* W, const float* bias, float* Z,
                    int M, int Kw, int N, int doStats) {
        int Kpad = (Kw + 3) & ~3;
        if (doStats) zero(stats, 4096);
        dim3 g((unsigned)((M + 15) / 16), (unsigned)((N + 15) / 16));
        pn2_gemm_kernel<<<g, 32, 2 * GEMM_KSTAGE * 16 * sizeof(float), stream>>>(
            X, W, bias, Z, stats, stats + 2048, M, Kpad, Kw, N, doStats);
    };
    auto bnrelu = [&](float* Z, const float* gamma, const float* beta,
                      int M, int N, int relu) {
        pn2_bn_relu_kernel<<<nblk((long)M * N), 256, 0, stream>>>(
            Z, stats, stats + 2048, gamma, beta, M, N, relu);
    };
    auto run_mlp = [&](const float* X0, int M, const int* dims, Lyr* layers) -> float* {
        const float* X = X0;
        float* bufs[2] = { bufA, bufB };
        int cur = 0; float* out = nullptr;
        for (int l = 0; l < 3; ++l) {
            float* Z = bufs[cur];
            gemm(X, layers[l].W, nullptr, Z, M, dims[l], dims[l + 1], 1);
            bnrelu(Z, layers[l].gamma, layers[l].beta, M, dims[l + 1], 1);
            X = Z; out = Z; cur ^= 1;
        }
        return out;
    };
    auto copy2d = [&](float* dst, long dp, const float* src, long sp, long rows, long cols) {
        pn2_copy2d_kernel<<<nblk(rows * cols), 256, 0, stream>>>(dst, dp, src, sp, rows, cols);
    };

    // ================= SA1: 16384 -> 512, r=0.02, ns=32, [3,32,32,64] =======
    {
        int S = 512, ns = 32, N = N0;
        pn2_fps_kernel<<<B, 256, (size_t)N * 4 + 256 * 8, stream>>>(pc, N, S, idx1);
        pn2_gather_points_kernel<<<nblk((long)B * S * 3), 256, 0, stream>>>(xyz1, pc, idx1, B, S, N, 3);
        pn2_ballquery_kernel<<<B * S, 32, 0, stream>>>(xyz1, pc, nidx1, B, S, N, ns, 0.0004f);
        pn2_group_kernel<<<nblk((long)B * S * ns * 4), 256, 0, stream>>>(pc, nullptr, xyz1, nidx1, bufB, B, S, N, ns, 0);
        static const int d1[4] = { 3, 32, 32, 64 };
        float* out = run_mlp(bufB, B * S * ns, d1, sa[0]);
        pn2_maxpool_kernel<<<nblk((long)B * S * 64), 256, 0, stream>>>(out, feats1, (long)B * S, ns, 64);
    }
    // ================= SA2: 512 -> 256, r=0.04, ns=16, [67,128,128,256] =====
    {
        int S = 256, ns = 16, N = 512, C = 64;
        pn2_fps_kernel<<<B, 256, (size_t)N * 4 + 256 * 8, stream>>>(xyz1, N, S, idx2);
        pn2_gather_points_kernel<<<nblk((long)B * S * 3), 256, 0, stream>>>(xyz2, xyz1, idx2, B, S, N, 3);
        pn2_ballquery_kernel<<<B * S, 32, 0, stream>>>(xyz2, xyz1, nidx2, B, S, N, ns, 0.0016f);
        pn2_group_kernel<<<nblk((long)B * S * ns * 68), 256, 0, stream>>>(xyz1, feats1, xyz2, nidx2, bufB, B, S, N, ns, C);
        static const int d2[4] = { 67, 128, 128, 256 };
        float* out = run_mlp(bufB, B * S * ns, d2, sa[1]);
        pn2_maxpool_kernel<<<nblk((long)B * S * 256), 256, 0, stream>>>(out, feats2, (long)B * S, ns, 256);
    }
    // ================= SA3: 256 -> 128, r=0.08, ns=16, [259,256,512,512] ====
    {
        int S = 128, ns = 16, N = 256, C = 256;
        pn2_fps_kernel<<<B, 256, (size_t)N * 4 + 256 * 8, stream>>>(xyz2, N, S, idx3);
        pn2_gather_points_kernel<<<nblk((long)B * S * 3), 256, 0, stream>>>(xyz3, xyz2, idx3, B, S, N, 3);
        pn2_ballquery_kernel<<<B * S, 32, 0, stream>>>(xyz3, xyz2, nidx3, B, S, N, ns, 0.0064f);
        pn2_group_kernel<<<nblk((long)B * S * ns * 260), 256, 0, stream>>>(xyz2, feats2, xyz3, nidx3, bufB, B, S, N, ns, C);
        static const int d3[4] = { 259, 256, 512, 512 };
        float* out = run_mlp(bufB, B * S * ns, d3, sa[2]);
        pn2_maxpool_kernel<<<nblk((long)B * S * 512), 256, 0, stream>>>(out, feats3, (long)B * S, ns, 512);
    }
    // ================= SA4: global, [515,512,1024,1024], max over 128 =======
    {
        int M = B * 128;                            // G pitch = 516 (pad col 515)
        zero(bufB, (long)M * 516);
        copy2d(bufB, 516, xyz3, 3, M, 3);           // G[:, :3]  = xyz
        copy2d(bufB + 3, 516, feats3, 512, M, 512); // G[:, 3:515] = features
        static const int d4[4] = { 515, 512, 1024, 1024 };
        float* out = run_mlp(bufB, M, d4, sa[3]);
        pn2_maxpool_kernel<<<nblk((long)B * 1024), 256, 0, stream>>>(out, feat4, (long)B, 128, 1024);
    }
    // ================= SA_SEP: FPS-4 of full cloud, [3,256,512,1024] ========
    {
        pn2_fps_kernel<<<B, 256, (size_t)N0 * 4 + 256 * 8, stream>>>(pc, N0, 4, idxs);
        pn2_gather_points_kernel<<<nblk((long)B * 4 * 3), 256, 0, stream>>>(sxyz, pc, idxs, B, 4, N0, 3);
        zero(Gs, (long)B * 4 * 4);                  // padded (B*4, 4) G
        copy2d(Gs, 4, sxyz, 3, (long)B * 4, 3);
        static const int ds[4] = { 3, 256, 512, 1024 };
        float* out = run_mlp(Gs, B * 4, ds, sa[4]);
        pn2_maxpool_kernel<<<nblk((long)B * 1024), 256, 0, stream>>>(out, sep, (long)B, 4, 1024);
    }
    // ================= FC head ==============================================
    copy2d(x1, 2048, feat4, 1024, B, 1024);
    copy2d(x1 + 1024, 2048, sep, 1024, B, 1024);
    gemm(x1, w1, b1, h1, B, 2048, 1024, 1);
    bnrelu(h1, g1, be1, B, 1024, 1);
    copy2d(x2, 2048, h1, 1024, B, 1024);
    copy2d(x2 + 1024, 2048, sep, 1024, B, 1024);
    gemm(x2, w2, b2, h2, B, 2048, 2048, 1);
    bnrelu(h2, g2, be2, B, 2048, 1);
    // (8,1,1024) -> transpose(0,2,1) is a flat no-op: write d_out directly.
    gemm(h2, w3, b3, (float*)d_out, B, 2048, 1024, 0);

    (void)in_sizes; (void)n_in; (void)out_size; (void)ws_size;
}